// GridPool_29832842838736
// MI455X (gfx1250) — compile-verified
//
#include <hip/hip_runtime.h>

#define CIN 48
#define COUT 96
#define GRIDSZ 0.05f
#define BN_EPS 1e-5f
#define TABLE (1 << 20)   // voxel-id table (8 * ~20^3 ~ 64K actual; 16x headroom)

typedef __attribute__((ext_vector_type(2))) float v2f;
typedef __attribute__((ext_vector_type(8))) float v8f;

// ---------------------------------------------------------------- utilities
__global__ void fill_u32_kernel(unsigned* p, long long count, unsigned value) {
  long long i = (long long)blockIdx.x * blockDim.x + threadIdx.x;
  long long stride = (long long)gridDim.x * blockDim.x;
  for (; i < count; i += stride) p[i] = value;
}

// Stage the block's contiguous 128x48 fp32 feature tile (24576 B) into LDS
// via CDNA5 async global->LDS copies (ASYNCcnt), fully coalesced b128s.
__device__ __forceinline__ void stage_feat_tile_async(const float* __restrict__ feat,
                                                      float* tile /*shared*/) {
  unsigned long long g = (unsigned long long)(feat) + threadIdx.x * 16ull;
  unsigned l = (unsigned)(size_t)tile + threadIdx.x * 16u;  // generic->LDS offset (low 32b)
#pragma unroll
  for (int k = 0; k < 6; ++k) {   // 256 thr * 16 B * 6 = 24576 B
    asm volatile("global_load_async_to_lds_b128 %0, %1, off"
                 :: "v"(l + (unsigned)(k * 4096)),
                    "v"(g + (unsigned long long)(k * 4096))
                 : "memory");
  }
  asm volatile("s_wait_asynccnt 0" ::: "memory");
}

// ------------------------------------------------ GEMM pass A: BN statistics
// y = feat @ W^T via V_WMMA_F32_16X16X4_F32; accumulate per-channel sum/sumsq.
__global__ void __launch_bounds__(256)
wmma_stats_kernel(const float* __restrict__ feat, const float* __restrict__ W,
                  float* __restrict__ sums, float* __restrict__ sumsq) {
  __shared__ __align__(16) float Wl[COUT * CIN];
  __shared__ __align__(16) float Ft[128 * CIN];
  for (int t = threadIdx.x; t < COUT * CIN; t += 256) Wl[t] = W[t];
  const float* gtile = feat + (size_t)blockIdx.x * 128 * CIN;
  __builtin_prefetch(gtile + 128 * CIN, 0, 0);            // warm next block's tile
  stage_feat_tile_async(gtile, Ft);
  __syncthreads();

  const int wave = threadIdx.x >> 5;
  const int lane = threadIdx.x & 31;
  const int half = lane >> 4;        // 0 => K pair {0,1}, 1 => K pair {2,3}
  const int lrow = lane & 15;
  const float* frow = &Ft[(wave * 16 + lrow) * CIN];

  v8f acc[6];
  for (int c = 0; c < 6; ++c)
    for (int e = 0; e < 8; ++e) acc[c][e] = 0.0f;

  for (int kt = 0; kt < CIN; kt += 4) {
    const int kk = kt + half * 2;
    v2f a; a.x = frow[kk]; a.y = frow[kk + 1];
#pragma unroll
    for (int ct = 0; ct < 6; ++ct) {
      const int col = ct * 16 + lrow;
      v2f b; b.x = Wl[col * CIN + kk]; b.y = Wl[col * CIN + kk + 1];
      acc[ct] = __builtin_amdgcn_wmma_f32_16x16x4_f32(
          false, a, false, b, (short)0, acc[ct], false, false);
    }
  }
  // per-column partial sums across this 16-row tile
  for (int ct = 0; ct < 6; ++ct) {
    float s = 0.0f, ss = 0.0f;
    for (int e = 0; e < 8; ++e) { float v = acc[ct][e]; s += v; ss += v * v; }
    s  += __shfl_xor(s, 16, 32);   // combine rows 0-7 with rows 8-15
    ss += __shfl_xor(ss, 16, 32);
    if (lane < 16) {
      atomicAdd(&sums[ct * 16 + lane], s);
      atomicAdd(&sumsq[ct * 16 + lane], ss);
    }
  }
}

__global__ void bn_finalize_kernel(const float* __restrict__ sums,
                                   const float* __restrict__ sumsq,
                                   const float* __restrict__ gamma,
                                   const float* __restrict__ beta,
                                   float* __restrict__ scale,
                                   float* __restrict__ shift, float fn) {
  int t = threadIdx.x;
  if (t < COUT) {
    float mu = sums[t] / fn;
    float var = sumsq[t] / fn - mu * mu;
    float sc = gamma[t] * rsqrtf(var + BN_EPS);
    scale[t] = sc;
    shift[t] = beta[t] - mu * sc;
  }
}

// --------------------------------------------- per-batch min coordinate
__global__ void __launch_bounds__(256)
coord_min_kernel(const float* __restrict__ coord, const int* __restrict__ batch,
                 unsigned* __restrict__ start_min, int n) {
  __shared__ unsigned red[3][256];
  int i = blockIdx.x * 256 + threadIdx.x;
  unsigned v0 = 0x7F7FFFFFu, v1 = 0x7F7FFFFFu, v2 = 0x7F7FFFFFu;
  if (i < n) {
    v0 = __float_as_uint(coord[i * 3 + 0]);
    v1 = __float_as_uint(coord[i * 3 + 1]);
    v2 = __float_as_uint(coord[i * 3 + 2]);
  }
  red[0][threadIdx.x] = v0; red[1][threadIdx.x] = v1; red[2][threadIdx.x] = v2;
  __syncthreads();
  for (int s = 128; s > 0; s >>= 1) {
    if (threadIdx.x < s) {
      for (int d = 0; d < 3; ++d) {
        unsigned a = red[d][threadIdx.x], b = red[d][threadIdx.x + s];
        red[d][threadIdx.x] = a < b ? a : b;
      }
    }
    __syncthreads();
  }
  if (threadIdx.x < 3 && blockIdx.x * 256 < n) {
    int b = batch[blockIdx.x * 256];   // N/B block-aligned: one batch per block
    atomicMin(&start_min[b * 3 + threadIdx.x], red[threadIdx.x][0]);
  }
}

// --------------------------------------------- voxel grid extents (nmax-1)
__global__ void __launch_bounds__(256)
grid_extent_kernel(const float* __restrict__ coord, const int* __restrict__ batch,
                   const unsigned* __restrict__ start_min, int* __restrict__ gmax,
                   int n) {
  __shared__ int red[3][256];
  int i = blockIdx.x * 256 + threadIdx.x;
  int g0 = 0, g1 = 0, g2 = 0;
  if (i < n) {
    int b = batch[i];
    g0 = (int)floorf((coord[i*3+0] - __uint_as_float(start_min[b*3+0])) / GRIDSZ);
    g1 = (int)floorf((coord[i*3+1] - __uint_as_float(start_min[b*3+1])) / GRIDSZ);
    g2 = (int)floorf((coord[i*3+2] - __uint_as_float(start_min[b*3+2])) / GRIDSZ);
  }
  red[0][threadIdx.x] = g0; red[1][threadIdx.x] = g1; red[2][threadIdx.x] = g2;
  __syncthreads();
  for (int s = 128; s > 0; s >>= 1) {
    if (threadIdx.x < s) {
      for (int d = 0; d < 3; ++d) {
        int a = red[d][threadIdx.x], b = red[d][threadIdx.x + s];
        red[d][threadIdx.x] = a > b ? a : b;
      }
    }
    __syncthreads();
  }
  if (threadIdx.x < 3) atomicMax(&gmax[threadIdx.x], red[threadIdx.x][0]);
}

// --------------------------------------------- voxel id + presence flags
__global__ void __launch_bounds__(256)
cluster_kernel(const float* __restrict__ coord, const int* __restrict__ batch,
               const unsigned* __restrict__ start_min, const int* __restrict__ gmax,
               int* __restrict__ cluster, int* __restrict__ flags, int n) {
  int i = blockIdx.x * 256 + threadIdx.x;
  if (i >= n) return;
  int nm0 = gmax[0] + 1, nm1 = gmax[1] + 1, nm2 = gmax[2] + 1;
  int b = batch[i];
  int g0 = (int)floorf((coord[i*3+0] - __uint_as_float(start_min[b*3+0])) / GRIDSZ);
  int g1 = (int)floorf((coord[i*3+1] - __uint_as_float(start_min[b*3+1])) / GRIDSZ);
  int g2 = (int)floorf((coord[i*3+2] - __uint_as_float(start_min[b*3+2])) / GRIDSZ);
  int cl = ((b * nm0 + g0) * nm1 + g1) * nm2 + g2;
  if (cl < 0) cl = 0;
  if (cl >= TABLE) cl = TABLE - 1;
  cluster[i] = cl;
  flags[cl] = 1;   // benign race: everyone writes 1
}

// --------------------------------------------- exclusive scan over flags
// Single workgroup, sequential chunks => ranks == sorted-unique inverse ids.
// shfl-based warp scan: 4 barriers per 1024-element chunk.
__global__ void __launch_bounds__(1024)
scan_kernel(const int* __restrict__ flags, int* __restrict__ ranks) {
  __shared__ int wsum[32];
  __shared__ int woff[32];
  __shared__ int ctot;
  __shared__ int running;
  const int tid = threadIdx.x, lane = tid & 31, wid = tid >> 5;
  if (tid == 0) running = 0;
  __syncthreads();
  for (int base = 0; base < TABLE; base += 1024) {
    int v = flags[base + tid];
    int incl = v;
#pragma unroll
    for (int off = 1; off < 32; off <<= 1) {
      int t = __shfl_up(incl, off, 32);
      if (lane >= off) incl += t;
    }
    if (lane == 31) wsum[wid] = incl;
    __syncthreads();
    if (wid == 0) {
      int s = wsum[lane];
      int si = s;
#pragma unroll
      for (int off = 1; off < 32; off <<= 1) {
        int t = __shfl_up(si, off, 32);
        if (lane >= off) si += t;
      }
      woff[lane] = si - s;           // exclusive warp offset
      if (lane == 31) ctot = si;     // chunk total
    }
    __syncthreads();
    ranks[base + tid] = running + woff[wid] + incl - v;   // exclusive scan
    __syncthreads();
    if (tid == 0) running += ctot;
    __syncthreads();
  }
}

// --------------------------------------------- segment scatter reductions
__global__ void __launch_bounds__(256)
scatter_kernel(const float* __restrict__ coord, const int* __restrict__ batch,
               const int* __restrict__ cluster, const int* __restrict__ ranks,
               float* __restrict__ coord_sum, float* __restrict__ counts,
               int* __restrict__ batch_min, int* __restrict__ inv_out, int n) {
  int i = blockIdx.x * 256 + threadIdx.x;
  if (i >= n) return;
  int r = ranks[cluster[i]];
  inv_out[i] = r;
  atomicAdd(&coord_sum[r * 3 + 0], coord[i * 3 + 0]);
  atomicAdd(&coord_sum[r * 3 + 1], coord[i * 3 + 1]);
  atomicAdd(&coord_sum[r * 3 + 2], coord[i * 3 + 2]);
  atomicAdd(&counts[r], 1.0f);
  atomicMin(&batch_min[r], batch[i]);
}

// ---------------- GEMM pass B: recompute y, BN+ReLU, segment-max into output
__global__ void __launch_bounds__(256)
wmma_apply_kernel(const float* __restrict__ feat, const float* __restrict__ W,
                  const float* __restrict__ scale, const float* __restrict__ shift,
                  const int* __restrict__ inv, unsigned* __restrict__ featmax) {
  __shared__ __align__(16) float Wl[COUT * CIN];
  __shared__ __align__(16) float Ft[128 * CIN];
  for (int t = threadIdx.x; t < COUT * CIN; t += 256) Wl[t] = W[t];
  const float* gtile = feat + (size_t)blockIdx.x * 128 * CIN;
  __builtin_prefetch(gtile + 128 * CIN, 0, 0);
  stage_feat_tile_async(gtile, Ft);
  __syncthreads();

  const int wave = threadIdx.x >> 5;
  const int lane = threadIdx.x & 31;
  const int half = lane >> 4;
  const int lrow = lane & 15;
  const int m0 = blockIdx.x * 128 + wave * 16;
  const float* frow = &Ft[(wave * 16 + lrow) * CIN];

  v8f acc[6];
  for (int c = 0; c < 6; ++c)
    for (int e = 0; e < 8; ++e) acc[c][e] = 0.0f;

  for (int kt = 0; kt < CIN; kt += 4) {
    const int kk = kt + half * 2;
    v2f a; a.x = frow[kk]; a.y = frow[kk + 1];
#pragma unroll
    for (int ct = 0; ct < 6; ++ct) {
      const int col = ct * 16 + lrow;
      v2f b; b.x = Wl[col * CIN + kk]; b.y = Wl[col * CIN + kk + 1];
      acc[ct] = __builtin_amdgcn_wmma_f32_16x16x4_f32(
          false, a, false, b, (short)0, acc[ct], false, false);
    }
  }

  int rloc[8];
  {
    const int mbase = m0 + half * 8;   // lanes 0-15: rows 0-7; lanes 16-31: rows 8-15
    for (int e = 0; e < 8; ++e) rloc[e] = inv[mbase + e];
  }
  for (int ct = 0; ct < 6; ++ct) {
    const int col = ct * 16 + lrow;
    const float sc = scale[col], sh = shift[col];
    for (int e = 0; e < 8; ++e) {
      float y = fmaxf(acc[ct][e] * sc + sh, 0.0f);   // ReLU => y >= 0
      atomicMax(&featmax[(size_t)rloc[e] * COUT + col], __float_as_uint(y));
    }
  }
}

// --------------------------------------------- finalize mean / fill values
__global__ void __launch_bounds__(256)
finalize_kernel(float* __restrict__ coord_out, const float* __restrict__ counts,
                int* __restrict__ batch_out, int n) {
  int r = blockIdx.x * 256 + threadIdx.x;
  if (r >= n) return;
  float c = counts[r];
  if (c > 0.0f) {
    float ic = 1.0f / c;
    coord_out[r * 3 + 0] *= ic;
    coord_out[r * 3 + 1] *= ic;
    coord_out[r * 3 + 2] *= ic;
  } else {
    batch_out[r] = 0;   // invalid slots: batch 0 (coord/feat already zero)
  }
}

extern "C" void kernel_launch(void* const* d_in, const int* in_sizes, int n_in,
                              void* d_out, int out_size, void* d_ws, size_t ws_size,
                              hipStream_t stream) {
  const float* coord = (const float*)d_in[0];
  const float* feat  = (const float*)d_in[1];
  const int*   batch = (const int*)d_in[2];
  const float* W     = (const float*)d_in[3];
  const float* gamma = (const float*)d_in[4];
  const float* beta  = (const float*)d_in[5];
  const int n = in_sizes[2];                      // number of points

  // d_out layout: coord_out[3n] | feat_out[96n] | batch_out[n] | inv[n]
  float* out_f     = (float*)d_out;
  float* coord_out = out_f;
  float* feat_out  = out_f + (size_t)3 * n;
  int*   batch_out = (int*)(out_f + (size_t)99 * n);
  int*   inv_out   = (int*)(out_f + (size_t)100 * n);

  // workspace layout (4-byte units)
  float* ws        = (float*)d_ws;
  float* sums      = ws;                   // 96
  float* sumsq     = ws + 96;              // 96
  float* scale     = ws + 192;             // 96
  float* shift     = ws + 288;             // 96
  unsigned* start_min = (unsigned*)(ws + 384);  // 8*3
  int*   gmax      = (int*)(ws + 408);     // 3 (+pad to 416)
  float* counts    = ws + 416;             // n
  int*   cluster   = (int*)(ws + 416) + n; // n
  int*   flags     = cluster + n;          // TABLE
  int*   ranks     = flags + TABLE;        // TABLE

  const int nb = (n + 255) / 256;

  // ---- init (d_out/ws are poisoned by harness; rebuild every call) ----
  fill_u32_kernel<<<2048, 256, 0, stream>>>((unsigned*)coord_out, (long long)99 * n, 0u);
  fill_u32_kernel<<<256, 256, 0, stream>>>((unsigned*)batch_out, n, 0x7FFFFFFFu);
  fill_u32_kernel<<<1, 256, 0, stream>>>((unsigned*)sums, 384, 0u);
  fill_u32_kernel<<<1, 32, 0, stream>>>(start_min, 24, 0x7F7FFFFFu);
  fill_u32_kernel<<<1, 32, 0, stream>>>((unsigned*)gmax, 8, 0u);
  fill_u32_kernel<<<256, 256, 0, stream>>>((unsigned*)counts, n, 0u);
  fill_u32_kernel<<<1024, 256, 0, stream>>>((unsigned*)flags, TABLE, 0u);

  // ---- pipeline (stream order provides dependencies) ----
  wmma_stats_kernel<<<n / 128, 256, 0, stream>>>(feat, W, sums, sumsq);
  bn_finalize_kernel<<<1, 128, 0, stream>>>(sums, sumsq, gamma, beta, scale, shift, (float)n);
  coord_min_kernel<<<nb, 256, 0, stream>>>(coord, batch, start_min, n);
  grid_extent_kernel<<<nb, 256, 0, stream>>>(coord, batch, start_min, gmax, n);
  cluster_kernel<<<nb, 256, 0, stream>>>(coord, batch, start_min, gmax, cluster, flags, n);
  scan_kernel<<<1, 1024, 0, stream>>>(flags, ranks);
  scatter_kernel<<<nb, 256, 0, stream>>>(coord, batch, cluster, ranks,
                                         coord_out, counts, batch_out, inv_out, n);
  wmma_apply_kernel<<<n / 128, 256, 0, stream>>>(feat, W, scale, shift, inv_out,
                                                 (unsigned*)feat_out);
  finalize_kernel<<<nb, 256, 0, stream>>>(coord_out, counts, batch_out, n);
}